// GraphLayer_42477226557665
// MI455X (gfx1250) — compile-verified
//
#include <hip/hip_runtime.h>

typedef float v2f __attribute__((ext_vector_type(2)));
typedef float v8f __attribute__((ext_vector_type(8)));

// ---------------------------------------------------------------------------
// Degree pipeline: deg (int) -> dinv = rsqrt(deg) in-place (same buffer)
// ---------------------------------------------------------------------------
__global__ void deg_init_kernel(int* __restrict__ deg, int n) {
    int i = blockIdx.x * blockDim.x + threadIdx.x;
    if (i < n) deg[i] = 1;                    // self-loop counts as 1
}

__global__ void deg_edges_kernel(const long long* __restrict__ ei, int* __restrict__ deg, int E) {
    int e = blockIdx.x * blockDim.x + threadIdx.x;
    if (e < E) {
        int d = (int)ei[(size_t)E + e];       // dst row of edge_index
        atomicAdd(&deg[d], 1);                // native int atomic
    }
}

__global__ void deg_to_dinv_kernel(float* __restrict__ buf, int n) {
    int i = blockIdx.x * blockDim.x + threadIdx.x;
    if (i < n) {
        int d = ((const int*)buf)[i];
        buf[i] = rsqrtf((float)d);
    }
}

// ---------------------------------------------------------------------------
// GEMM via V_WMMA_F32_16X16X4_F32.
// One wave computes a 16-row x 64-col strip of hx = x @ W, and also writes the
// self-loop seeded accumulator agg = hx * dinv^2.
// NOTE: x and agg may alias (each wave writes only rows it alone reads, and no
// __restrict__ on them keeps all A-loads ordered before the stores).
// ---------------------------------------------------------------------------
__global__ void gcn_gemm_wmma_kernel(const float* x, const float* __restrict__ w,
                                     const float* __restrict__ dinv,
                                     float* __restrict__ hx, float* agg, int nrows) {
    const int lane = threadIdx.x & 31;
    const int wave = threadIdx.x >> 5;
    const int row0 = (blockIdx.x * 8 + wave) * 16;
    if (row0 >= nrows) return;                 // wave-uniform: EXEC stays all-1s

    const int m  = lane & 15;                  // M (for A) / N (for B,C,D)
    const int kp = (lane >> 4) << 1;           // K sub-pair: 0 or 2

    v8f acc0 = {}, acc1 = {}, acc2 = {}, acc3 = {};

    for (int k0 = 0; k0 < 64; k0 += 4) {
        // A tile 16x4: lane m<16 holds K=k0+{0,1}; lanes 16-31 hold K=k0+{2,3}
        const float* ap = x + (size_t)(row0 + m) * 64 + (k0 + kp);
        v2f a; a.x = ap[0]; a.y = ap[1];
        // B tiles 4x16 per 16-col group: v0 = W[k0+kp][n], v1 = W[k0+kp+1][n]
        const float* bp = w + (size_t)(k0 + kp) * 64 + m;
        v2f b0; b0.x = bp[0];  b0.y = bp[64];
        v2f b1; b1.x = bp[16]; b1.y = bp[80];
        v2f b2; b2.x = bp[32]; b2.y = bp[96];
        v2f b3; b3.x = bp[48]; b3.y = bp[112];
        acc0 = __builtin_amdgcn_wmma_f32_16x16x4_f32(false, a, false, b0, (short)0, acc0, false, false);
        acc1 = __builtin_amdgcn_wmma_f32_16x16x4_f32(false, a, false, b1, (short)0, acc1, false, false);
        acc2 = __builtin_amdgcn_wmma_f32_16x16x4_f32(false, a, false, b2, (short)0, acc2, false, false);
        acc3 = __builtin_amdgcn_wmma_f32_16x16x4_f32(false, a, false, b3, (short)0, acc3, false, false);
    }

    // C/D layout: element v of acc in this lane is row (row0 + v + 8*(lane>>4)), col group + m
    const int mh = (lane >> 4) << 3;
#pragma unroll
    for (int v = 0; v < 8; ++v) {
        const int row = row0 + mh + v;
        const float di = dinv[row];
        const float sc = di * di;
        const size_t base = (size_t)row * 64 + m;
        float c0 = acc0[v], c1 = acc1[v], c2 = acc2[v], c3 = acc3[v];
        hx[base +  0] = c0;  hx[base + 16] = c1;
        hx[base + 32] = c2;  hx[base + 48] = c3;
        agg[base +  0] = c0 * sc;  agg[base + 16] = c1 * sc;
        agg[base + 32] = c2 * sc;  agg[base + 48] = c3 * sc;
    }
}

// ---------------------------------------------------------------------------
// Edge scatter: 16 lanes per edge, each lane gathers a float4 of hx[src] and
// does 4 hardware f32 atomic adds into agg[dst]. hx/agg are L2-resident (20.5MB).
// ---------------------------------------------------------------------------
__global__ void edge_scatter_kernel(const long long* __restrict__ ei,
                                    const float* __restrict__ hx,
                                    const float* __restrict__ dinv,
                                    float* __restrict__ agg, int E) {
    int t = blockIdx.x * blockDim.x + threadIdx.x;
    int e = t >> 4;
    if (e >= E) return;
    const int f4 = (t & 15) << 2;
    const int s = (int)ei[e];
    const int d = (int)ei[(size_t)E + e];
    const float coef = dinv[s] * dinv[d];
    const float4 v = *(const float4*)(hx + (size_t)s * 64 + f4);
    float* ap = agg + (size_t)d * 64 + f4;
    unsafeAtomicAdd(ap + 0, v.x * coef);
    unsafeAtomicAdd(ap + 1, v.y * coef);
    unsafeAtomicAdd(ap + 2, v.z * coef);
    unsafeAtomicAdd(ap + 3, v.w * coef);
}

// ---------------------------------------------------------------------------
// Finalize: agg = (agg + b) [ReLU optional], float4-vectorized, in place.
// ---------------------------------------------------------------------------
__global__ void finalize_kernel(float* __restrict__ agg, const float* __restrict__ b,
                                int ngroups, int doRelu) {
    int g = blockIdx.x * blockDim.x + threadIdx.x;
    if (g >= ngroups) return;
    const int fb = (g & 15) << 2;
    float4 v = ((float4*)agg)[g];
    v.x += b[fb + 0]; v.y += b[fb + 1]; v.z += b[fb + 2]; v.w += b[fb + 3];
    if (doRelu) {
        v.x = fmaxf(v.x, 0.f); v.y = fmaxf(v.y, 0.f);
        v.z = fmaxf(v.z, 0.f); v.w = fmaxf(v.w, 0.f);
    }
    ((float4*)agg)[g] = v;
}

// ---------------------------------------------------------------------------
extern "C" void kernel_launch(void* const* d_in, const int* in_sizes, int n_in,
                              void* d_out, int out_size, void* d_ws, size_t ws_size,
                              hipStream_t stream) {
    const float*     X  = (const float*)d_in[0];
    const long long* ei = (const long long*)d_in[1];
    const float*     W1 = (const float*)d_in[2];
    const float*     b1 = (const float*)d_in[3];
    const float*     W2 = (const float*)d_in[4];
    const float*     b2 = (const float*)d_in[5];
    const float*     W3 = (const float*)d_in[6];
    const float*     b3 = (const float*)d_in[7];

    const int N = in_sizes[0] / 64;     // 80000
    const int E = in_sizes[1] / 2;      // 1280000
    float* out = (float*)d_out;

    // Workspace layout: dinv[N] (256-float aligned) | bufP[N*64] | bufQ[N*64]
    float* dinv = (float*)d_ws;
    float* bufP = dinv + (size_t)((N + 255) & ~255);
    float* bufQ = bufP + (size_t)N * 64;

    const int T = 256;
    const int gN   = (N + T - 1) / T;
    const int gE   = (E + T - 1) / T;
    const int gE16 = (E * 16 + T - 1) / T;          // 16 lanes per edge
    const int gFin = (N * 16 + T - 1) / T;          // float4 groups
    const int gGemm = (N / 16 + 7) / 8;             // 8 waves/block, 16 rows/wave

    // Degree -> dinv (stored in-place in dinv buffer)
    deg_init_kernel   <<<gN, T, 0, stream>>>((int*)dinv, N);
    deg_edges_kernel  <<<gE, T, 0, stream>>>(ei, (int*)dinv, E);
    deg_to_dinv_kernel<<<gN, T, 0, stream>>>(dinv, N);

    // Layer 1: x = X, hx = bufQ, agg = bufP
    gcn_gemm_wmma_kernel<<<gGemm, T, 0, stream>>>(X, W1, dinv, bufQ, bufP, N);
    edge_scatter_kernel <<<gE16, T, 0, stream>>>(ei, bufQ, dinv, bufP, E);
    finalize_kernel     <<<gFin, T, 0, stream>>>(bufP, b1, N * 16, 1);

    // Layer 2: x = bufP (agg aliases it safely), hx = bufQ
    gcn_gemm_wmma_kernel<<<gGemm, T, 0, stream>>>(bufP, W2, dinv, bufQ, bufP, N);
    edge_scatter_kernel <<<gE16, T, 0, stream>>>(ei, bufQ, dinv, bufP, E);
    finalize_kernel     <<<gFin, T, 0, stream>>>(bufP, b2, N * 16, 1);

    // Layer 3: accumulate straight into d_out (gemm epilogue fully initializes it)
    gcn_gemm_wmma_kernel<<<gGemm, T, 0, stream>>>(bufP, W3, dinv, bufQ, out, N);
    edge_scatter_kernel <<<gE16, T, 0, stream>>>(ei, bufQ, dinv, out, E);
    finalize_kernel     <<<gFin, T, 0, stream>>>(out, b3, N * 16, 0);
}